// EdgeAwareGATConv_20633022890437
// MI455X (gfx1250) — compile-verified
//
#include <hip/hip_runtime.h>
#include <hip/hip_bf16.h>

// ---------- constants from the reference ----------
#define HC 128      // H*C
#define HH 4        // heads
#define CC 32       // channels per head
#define K_IN 128    // SRC_IN / DST_IN
#define E_IN 16     // EDGE_IN
#define NEG_SLOPE 0.2f
#define EPS_F 1e-16f

typedef __attribute__((ext_vector_type(2))) float v2f;
typedef __attribute__((ext_vector_type(8))) float v8f;

// ---- ordered-int encoding for float atomic max (works for negatives) ----
__device__ __forceinline__ int f2ord(float f) {
    int i = __float_as_int(f);
    return i >= 0 ? i : (i ^ 0x7fffffff);
}
__device__ __forceinline__ float ord2f(int i) {
    return __int_as_float(i >= 0 ? i : (i ^ 0x7fffffff));
}

// ============================================================
// 0) Fold W_dst x a_dst -> u_dst[4][128], W_edge x a_edge -> v_edge[4][16]
// ============================================================
__global__ void prep_kernel(const float* __restrict__ Wd, const float* __restrict__ We,
                            const float* __restrict__ att,
                            float* __restrict__ u_dst, float* __restrict__ v_edge) {
    for (int idx = threadIdx.x; idx < HH * K_IN; idx += blockDim.x) {
        int h = idx >> 7, k = idx & (K_IN - 1);
        const float* a = att + h * (3 * CC);          // a_dst = att[h][0:32]
        float s = 0.f;
        #pragma unroll
        for (int c = 0; c < CC; ++c) s += Wd[(h * CC + c) * K_IN + k] * a[c];
        u_dst[idx] = s;
    }
    for (int idx = threadIdx.x; idx < HH * E_IN; idx += blockDim.x) {
        int h = idx >> 4, k = idx & (E_IN - 1);
        const float* a = att + h * (3 * CC) + 2 * CC; // a_edge = att[h][64:96]
        float s = 0.f;
        #pragma unroll
        for (int c = 0; c < CC; ++c) s += We[(h * CC + c) * E_IN + k] * a[c];
        v_edge[idx] = s;
    }
}

// ============================================================
// 1) src_p = x_src @ W_src^T  via V_WMMA_F32_16X16X4_F32
//    one wave -> one 16x16 tile; block = 8 waves = full 128-col stripe
// ============================================================
__global__ __launch_bounds__(256) void gemm_src_kernel(const float* __restrict__ X,
                                                       const float* __restrict__ W,
                                                       float* __restrict__ P) {
    const int lane = threadIdx.x & 31;
    const int wave = threadIdx.x >> 5;
    const int m0 = blockIdx.x << 4;          // 16-row tile
    const int col = (wave << 4) + (lane & 15);   // output channel (B column)
    const int row = m0 + (lane & 15);            // A row
    const int khalf = (lane >> 4) << 1;          // 0 or 2 (K sub-offset per ISA layout)

    const float* xrow = X + row * K_IN + khalf;
    const float* wrow = W + col * K_IN + khalf;  // B[k][col] = W[col][k]

    v8f acc = {};
    #pragma unroll
    for (int k0 = 0; k0 < K_IN; k0 += 4) {
        v2f a = *(const v2f*)(xrow + k0);
        v2f b = *(const v2f*)(wrow + k0);
        acc = __builtin_amdgcn_wmma_f32_16x16x4_f32(false, a, false, b,
                                                    (short)0, acc, false, false);
    }
    const int rbase = m0 + ((lane >> 4) << 3);   // VGPR j -> row rbase + j
    #pragma unroll
    for (int j = 0; j < 8; ++j)
        P[(rbase + j) * HC + col] = acc[j];
}

// ============================================================
// 2) per-node: alpha_src, alpha_dst; init smax/denom; zero agg output
//    one thread per (node, head)
// ============================================================
__global__ void node_alpha_kernel(const float* __restrict__ Xd, const float* __restrict__ src_p,
                                  const float* __restrict__ att, const float* __restrict__ u_dst,
                                  float* __restrict__ alpha_src, float* __restrict__ alpha_dst,
                                  int* __restrict__ smax, float* __restrict__ denom,
                                  float* __restrict__ agg, int n_src, int n_dst) {
    int i = blockIdx.x * blockDim.x + threadIdx.x;
    int n = i >> 2, h = i & 3;
    if (n < n_src) {
        const float* a = att + h * (3 * CC) + CC;   // a_src = att[h][32:64]
        const float* p = src_p + n * HC + h * CC;
        float s = 0.f;
        #pragma unroll
        for (int c = 0; c < CC; ++c) s += p[c] * a[c];
        alpha_src[i] = s;
    }
    if (n < n_dst) {
        const float* u = u_dst + h * K_IN;
        const float* x = Xd + n * K_IN;
        float s = 0.f;
        #pragma unroll
        for (int c = 0; c < K_IN; ++c) s += x[c] * u[c];
        alpha_dst[i] = s;
        smax[i] = (int)0x80000000;   // -> -inf under ordered encoding
        denom[i] = 0.f;
        float* ag = agg + (long long)n * HC + h * CC;
        #pragma unroll
        for (int c = 0; c < CC; ++c) ag[c] = 0.f;
    }
}

// ============================================================
// 3) per-edge: scores (leaky-relu'd) + segment max via ordered-int atomicMax
// ============================================================
__global__ void edge_score_kernel(const float* __restrict__ edge_attr, const int* __restrict__ ei,
                                  const float* __restrict__ v_edge,
                                  const float* __restrict__ alpha_src, const float* __restrict__ alpha_dst,
                                  float* __restrict__ scores, int* __restrict__ smax, int E) {
    int e = blockIdx.x * blockDim.x + threadIdx.x;
    if (e >= E) return;
    int s = ei[e];          // edge_index[0] = src
    int d = ei[E + e];      // edge_index[1] = dst
    float ea[E_IN];
    #pragma unroll
    for (int k = 0; k < E_IN; ++k) ea[k] = edge_attr[e * E_IN + k];
    #pragma unroll
    for (int h = 0; h < HH; ++h) {
        float ae = 0.f;
        #pragma unroll
        for (int k = 0; k < E_IN; ++k) ae += ea[k] * v_edge[h * E_IN + k];
        float sc = alpha_dst[d * 4 + h] + alpha_src[s * 4 + h] + ae;
        sc = sc > 0.f ? sc : sc * NEG_SLOPE;
        scores[e * 4 + h] = sc;
        atomicMax(&smax[d * 4 + h], f2ord(sc));
    }
}

// ============================================================
// 4) per-(edge,head): ex = exp(score - smax[dst]); denom += ex (in-place buf)
// ============================================================
__global__ void edge_exp_kernel(const int* __restrict__ ei, float* __restrict__ sc_ex,
                                const int* __restrict__ smax, float* __restrict__ denom, int E) {
    int i = blockIdx.x * blockDim.x + threadIdx.x;
    if (i >= E * 4) return;
    int e = i >> 2, h = i & 3;
    int d = ei[E + e];
    float m = ord2f(smax[d * 4 + h]);
    float ex = __expf(sc_ex[i] - m);
    sc_ex[i] = ex;
    atomicAdd(&denom[d * 4 + h], ex);
}

// ============================================================
// 5) per-edge: attn out + message scatter-add (128 threads per edge)
// ============================================================
__global__ __launch_bounds__(256) void edge_agg_kernel(const int* __restrict__ ei,
                                                       const float* __restrict__ ex,
                                                       const float* __restrict__ denom,
                                                       const float* __restrict__ src_p,
                                                       float* __restrict__ agg,
                                                       float* __restrict__ attn_out, int E) {
    int tid = threadIdx.x;
    long long e = (long long)blockIdx.x * 2 + (tid >> 7);
    if (e >= E) return;
    int c = tid & (HC - 1);
    int h = c >> 5;
    int s = ei[e];
    int d = ei[E + e];
    float exv = ex[e * 4 + h];
    float a = exv / (denom[d * 4 + h] + EPS_F);
    if ((c & 31) == 0) attn_out[e * 4 + h] = a;
    float msg = src_p[s * HC + c] * a;
    atomicAdd(&agg[(long long)d * HC + c], msg);
}

// ============================================================
extern "C" void kernel_launch(void* const* d_in, const int* in_sizes, int n_in,
                              void* d_out, int out_size, void* d_ws, size_t ws_size,
                              hipStream_t stream) {
    const float* x_src    = (const float*)d_in[0];
    const float* x_dst    = (const float*)d_in[1];
    const float* edge_attr= (const float*)d_in[2];
    const float* W_src    = (const float*)d_in[3];
    const float* W_dst    = (const float*)d_in[4];
    const float* W_edge   = (const float*)d_in[5];
    const float* att_vec  = (const float*)d_in[6];
    const int*   ei       = (const int*)d_in[7];

    const int n_src = in_sizes[0] / K_IN;
    const int n_dst = in_sizes[1] / K_IN;
    const int E     = in_sizes[2] / E_IN;

    // output: [agg (n_dst*128)] [attn (E*4)]
    float* agg  = (float*)d_out;
    float* attn = (float*)d_out + (long long)n_dst * HC;

    // workspace layout (floats)
    float* ws = (float*)d_ws;
    float* src_p     = ws;                         ws += (long long)n_src * HC;
    float* alpha_src = ws;                         ws += (long long)n_src * HH;
    float* alpha_dst = ws;                         ws += (long long)n_dst * HH;
    int*   smax      = (int*)ws;                   ws += (long long)n_dst * HH;
    float* denom     = ws;                         ws += (long long)n_dst * HH;
    float* u_dst     = ws;                         ws += HH * K_IN;
    float* v_edge    = ws;                         ws += HH * E_IN;
    float* scores    = ws;                         ws += (long long)E * HH;

    prep_kernel<<<1, 128, 0, stream>>>(W_dst, W_edge, att_vec, u_dst, v_edge);

    gemm_src_kernel<<<n_src / 16, 256, 0, stream>>>(x_src, W_src, src_p);

    int nodes4 = ((n_src > n_dst ? n_src : n_dst) * HH + 255) / 256;
    node_alpha_kernel<<<nodes4, 256, 0, stream>>>(x_dst, src_p, att_vec, u_dst,
                                                  alpha_src, alpha_dst, smax, denom,
                                                  agg, n_src, n_dst);

    edge_score_kernel<<<(E + 255) / 256, 256, 0, stream>>>(edge_attr, ei, v_edge,
                                                           alpha_src, alpha_dst,
                                                           scores, smax, E);

    edge_exp_kernel<<<(E * 4 + 255) / 256, 256, 0, stream>>>(ei, scores, smax, denom, E);

    edge_agg_kernel<<<(E + 1) / 2, 256, 0, stream>>>(ei, scores, denom, src_p,
                                                     agg, attn, E);
}